// RNAOptimizationModel_55611236549006
// MI455X (gfx1250) — compile-verified
//
#include <hip/hip_runtime.h>
#include <math.h>

// ---------------------------------------------------------------------------
// Problem constants (from reference)
// ---------------------------------------------------------------------------
#define N_UNITS 8
#define M_ATOMS 1024
#define N_PAIRS 28                        // 8*7/2
#define ROW_TILES (M_ATOMS / 16)          // 64
#define PAIR_BLOCKS (N_PAIRS * ROW_TILES) // 1792

typedef __attribute__((ext_vector_type(2))) float v2f;
typedef __attribute__((ext_vector_type(4))) float v4f;
typedef __attribute__((ext_vector_type(8))) float v8f;

// ---------------------------------------------------------------------------
// Euler (ZYX: Rz*Ry*Rx) rotation matrix, row-major R[9]
// ---------------------------------------------------------------------------
__device__ inline void euler_R(float phi, float th, float psi, float R[9]) {
    float cph = cosf(phi), sph = sinf(phi);
    float cth = cosf(th),  sth = sinf(th);
    float cps = cosf(psi), sps = sinf(psi);
    // A = Rz*Ry
    float A00 =  cps * cth, A01 = -sps, A02 = cps * sth;
    float A10 =  sps * cth, A11 =  cps, A12 = sps * sth;
    float A20 = -sth,       A21 = 0.f,  A22 = cth;
    // R = A*Rx,  Rx = [[1,0,0],[0,cph,-sph],[0,sph,cph]]
    R[0] = A00; R[1] = A01 * cph + A02 * sph; R[2] = -A01 * sph + A02 * cph;
    R[3] = A10; R[4] = A11 * cph + A12 * sph; R[5] = -A11 * sph + A12 * cph;
    R[6] = A20; R[7] = A21 * cph + A22 * sph; R[8] = -A21 * sph + A22 * cph;
}

// ---------------------------------------------------------------------------
// Kernel 1: transform coordinates per unit; write the WMMA-ready packed panel
//   PB[n][m][4] = {x, y, z, |t|^2}   (b64-loadable per-lane operand slices)
//   SQ[n][m]    = |t|^2              (b128 broadcast loads for the C operand)
// plus the per-unit sum of |t|^2 (for L1). 8 blocks x 256 threads.
// ---------------------------------------------------------------------------
__global__ void __launch_bounds__(256)
setup_kernel(const float* __restrict__ pos,
             const float* __restrict__ euler,
             const float* __restrict__ coords,
             float* __restrict__ PB, float* __restrict__ SQ,
             float* __restrict__ sqsum) {
    const int n = blockIdx.x;
    const int tid = threadIdx.x;
    float R[9];
    euler_R(euler[n * 3 + 0], euler[n * 3 + 1], euler[n * 3 + 2], R);
    const float px = pos[n * 3 + 0], py = pos[n * 3 + 1], pz = pos[n * 3 + 2];

    v4f* pb = reinterpret_cast<v4f*>(PB) + (size_t)n * M_ATOMS;

    float local = 0.0f;
    for (int m = tid; m < M_ATOMS; m += 256) {
        float cx = coords[m * 3 + 0], cy = coords[m * 3 + 1], cz = coords[m * 3 + 2];
        float x = R[0] * cx + R[1] * cy + R[2] * cz + px;
        float y = R[3] * cx + R[4] * cy + R[5] * cz + py;
        float z = R[6] * cx + R[7] * cy + R[8] * cz + pz;
        float sq = x * x + y * y + z * z;
        v4f v; v.x = x; v.y = y; v.z = z; v.w = sq;
        pb[m] = v;                       // global_store_b128
        SQ[n * M_ATOMS + m] = sq;
        local += sq;
    }
    __shared__ float red[256];
    red[tid] = local;
    __syncthreads();
    for (int off = 128; off > 0; off >>= 1) {
        if (tid < off) red[tid] += red[tid + off];
        __syncthreads();
    }
    if (tid == 0) sqsum[n] = red[0];
}

// ---------------------------------------------------------------------------
// Kernel 2: small terms — L_com and box-overlap L_overlap. 1 block x 128.
// ---------------------------------------------------------------------------
__global__ void small_terms_kernel(const float* __restrict__ pos,
                                   const float* __restrict__ euler,
                                   const float* __restrict__ coords,
                                   float* __restrict__ out_lcom,
                                   float* __restrict__ out_lov) {
    __shared__ float smn[3][128];
    __shared__ float smx[3][128];
    const int tid = threadIdx.x;
    float mn[3] = {3.0e38f, 3.0e38f, 3.0e38f};
    float mx[3] = {-3.0e38f, -3.0e38f, -3.0e38f};
    for (int m = tid; m < M_ATOMS; m += 128) {
        for (int d = 0; d < 3; ++d) {
            float v = coords[m * 3 + d];
            mn[d] = fminf(mn[d], v);
            mx[d] = fmaxf(mx[d], v);
        }
    }
    for (int d = 0; d < 3; ++d) { smn[d][tid] = mn[d]; smx[d][tid] = mx[d]; }
    __syncthreads();
    for (int off = 64; off > 0; off >>= 1) {
        if (tid < off) {
            for (int d = 0; d < 3; ++d) {
                smn[d][tid] = fminf(smn[d][tid], smn[d][tid + off]);
                smx[d][tid] = fmaxf(smx[d][tid], smx[d][tid + off]);
            }
        }
        __syncthreads();
    }
    if (tid == 0) {
        float bmin[3], bmax[3];
        for (int d = 0; d < 3; ++d) { bmin[d] = smn[d][0]; bmax[d] = smx[d][0]; }

        // L_com = ALPHA * |sum_n pos_n|^2
        float com[3] = {0.f, 0.f, 0.f};
        for (int n = 0; n < N_UNITS; ++n)
            for (int d = 0; d < 3; ++d) com[d] += pos[n * 3 + d];
        *out_lcom = com[0] * com[0] + com[1] * com[1] + com[2] * com[2]; // ALPHA=1

        // boxes[n,k] = box[k] + (R_n @ box[k] + pos_n)
        float mins[N_UNITS][3], maxs[N_UNITS][3];
        for (int n = 0; n < N_UNITS; ++n) {
            float R[9];
            euler_R(euler[n * 3 + 0], euler[n * 3 + 1], euler[n * 3 + 2], R);
            for (int d = 0; d < 3; ++d) {
                float t0 = R[d * 3 + 0] * bmin[0] + R[d * 3 + 1] * bmin[1] +
                           R[d * 3 + 2] * bmin[2] + pos[n * 3 + d];
                float t1 = R[d * 3 + 0] * bmax[0] + R[d * 3 + 1] * bmax[1] +
                           R[d * 3 + 2] * bmax[2] + pos[n * 3 + d];
                mins[n][d] = bmin[d] + t0;
                maxs[n][d] = bmax[d] + t1;
            }
        }
        float ov = 0.0f;
        for (int i = 0; i < N_UNITS; ++i)
            for (int j = i + 1; j < N_UNITS; ++j) {
                float vol = 1.0f;
                for (int d = 0; d < 3; ++d) {
                    float lo = fmaxf(mins[i][d], mins[j][d]);
                    float hi = fminf(maxs[i][d], maxs[j][d]);
                    float e = hi - lo;
                    vol *= (e > 0.0f) ? e : 0.0f;
                }
                ov += vol;
            }
        *out_lov = ov * (10.0f * 3.0f); // LAMBDA2 * 3
    }
}

// ---------------------------------------------------------------------------
// Kernel 3: WMMA pair-interaction kernel.
//   D = A x B + C per V_WMMA_F32_16X16X4_F32 with
//     A[m]   = [-2x, -2y, -2z, 1]   (unit i)
//     B[:,n] = [x', y', z', sq']    (unit j)
//     C[m,n] = sq_i[m]
//   => D = full 16x16 tile of squared distances d2.
// Grid: 28 pairs * 64 row-tiles, 256 threads (8 waves), each wave does
// 8 fully-unrolled column tiles (8 static WMMAs). Per-block partial sums go
// to unique slots — no atomics, deterministic.
// ---------------------------------------------------------------------------
__global__ void __launch_bounds__(256)
pair_pen_kernel(const float* __restrict__ PB,
                const float* __restrict__ SQ,
                float* __restrict__ partial) {
    const int tid = threadIdx.x;
    const int wave = tid >> 5;
    const int lane = tid & 31;
    const int half = lane >> 4;   // K in {0,1} (lanes 0-15) vs {2,3} (lanes 16-31)
    const int lr = lane & 15;     // row (A) / col (B) within the 16x16 tile

    const int blk = blockIdx.x;
    const int pair = blk >> 6;    // / ROW_TILES
    const int row_tile = blk & 63;

    // linear pair -> (i, j), i < j  (uniform scalar loop)
    int p = pair, i = 0, span = N_UNITS - 1;
    while (p >= span) { p -= span; ++i; --span; }
    const int j = i + 1 + p;

    // v2f views of the packed panels: element index = m*2 + half
    const v2f* pbi = reinterpret_cast<const v2f*>(PB) + (size_t)i * M_ATOMS * 2;
    const v2f* pbj = reinterpret_cast<const v2f*>(PB) + (size_t)j * M_ATOMS * 2;
    const float* sqi = SQ + i * M_ATOMS;

    __builtin_prefetch(pbj, 0, 3);        // global_prefetch_b8
    __builtin_prefetch(pbi, 0, 3);

    // --- A operand (2 VGPRs): one b64 load from unit-i panel.
    //     half0 lane gets {x,y} -> {-2x,-2y}=K0,K1 ; half1 gets {z,sq} -> {-2z,1}=K2,K3
    const int row0 = row_tile * 16;
    const v2f av = pbi[(row0 + lr) * 2 + half];
    v2f a;
    a.x = -2.0f * av.x;
    a.y = half ? 1.0f : (-2.0f * av.y);

    // --- C operand (8 VGPRs): C[m][n] = sq_i[m]; VGPR v holds M = v + 8*half.
    //     Two b128 broadcast loads of 8 contiguous floats.
    const v4f* cp = reinterpret_cast<const v4f*>(sqi + row0 + 8 * half);
    const v4f c0 = cp[0], c1 = cp[1];
    v8f c;
    c[0] = c0.x; c[1] = c0.y; c[2] = c0.z; c[3] = c0.w;
    c[4] = c1.x; c[5] = c1.y; c[6] = c1.z; c[7] = c1.w;

    float s = 0.0f;
#pragma unroll
    for (int t = 0; t < 8; ++t) {
        const int ct = wave + t * 8;      // column tile; trip count uniform
        // --- B operand (2 VGPRs): one b64 load from unit-j panel.
        const v2f b = pbj[(ct * 16 + lr) * 2 + half];

        // D[m][n] = sq_i[m] + sq_j[n] - 2 * t_i[m].t_j[n] = d2
        v8f d = __builtin_amdgcn_wmma_f32_16x16x4_f32(
            false, a, false, b, (short)0, c, false, false);

        // pen = exp(-(d2 - DELTA^2)/SIGMA^2) = exp(1 - d2)
#pragma unroll
        for (int v = 0; v < 8; ++v) s += __expf(1.0f - d[v]);
    }

    __shared__ float red[256];
    red[tid] = s;
    __syncthreads();
    for (int off = 128; off > 0; off >>= 1) {
        if (tid < off) red[tid] += red[tid + off];
        __syncthreads();
    }
    if (tid == 0) partial[blk] = red[0];
}

// ---------------------------------------------------------------------------
// Kernel 4: deterministic final combine (fixed summation order, 1 thread).
// ---------------------------------------------------------------------------
__global__ void finalize_kernel(const float* __restrict__ sqsum,
                                const float* __restrict__ partial,
                                const float* __restrict__ lcom,
                                const float* __restrict__ lov,
                                float* __restrict__ out) {
    float L1 = 0.0f;
    for (int n = 0; n < N_UNITS; ++n) L1 += sqsum[n];
    L1 *= (1.0f / (float)N_UNITS);
    float pen = 0.0f;
    for (int b = 0; b < PAIR_BLOCKS; ++b) pen += partial[b];
    const float Lint =
        10.0f * pen / ((float)N_UNITS * (N_UNITS - 1) * (float)M_ATOMS * (float)M_ATOMS);
    out[0] = L1 + lcom[0] + Lint + lov[0];
}

// ---------------------------------------------------------------------------
// Host-side launcher
// ---------------------------------------------------------------------------
extern "C" void kernel_launch(void* const* d_in, const int* in_sizes, int n_in,
                              void* d_out, int out_size, void* d_ws, size_t ws_size,
                              hipStream_t stream) {
    const float* pos    = (const float*)d_in[0]; // (8,3)
    const float* euler  = (const float*)d_in[1]; // (8,3)
    const float* coords = (const float*)d_in[2]; // (1024,3)
    float* out = (float*)d_out;

    float* ws = (float*)d_ws;
    float* lcom    = ws + 2;
    float* lov     = ws + 3;
    float* sqsum   = ws + 16;                 // 8 floats
    float* partial = ws + 32;                 // 1792 floats
    float* SQ = ws + 4096;                    // 8*1024 floats (16B aligned)
    float* PB = ws + 16384;                   // 8*1024*4 floats {x,y,z,sq}

    setup_kernel<<<N_UNITS, 256, 0, stream>>>(pos, euler, coords, PB, SQ, sqsum);
    small_terms_kernel<<<1, 128, 0, stream>>>(pos, euler, coords, lcom, lov);
    pair_pen_kernel<<<PAIR_BLOCKS, 256, 0, stream>>>(PB, SQ, partial);
    finalize_kernel<<<1, 1, 0, stream>>>(sqsum, partial, lcom, lov, out);
}